// Attention_Expert_11098195493449
// MI455X (gfx1250) — compile-verified
//
#include <hip/hip_runtime.h>
#include <hip/hip_bf16.h>

// ---------------------------------------------------------------------------
// XCA-style channel attention, FP32 throughout (memory-bound: ~40 GFLOP vs
// >1GB HBM traffic @23.3TB/s -> keep full precision, use f32 WMMA for GEMMs).
//
// Pipeline:
//   1) gemm1x1_wmma      : qkv = w_qkv @ x        (M=576, K=192, N=16384 / b)
//   2) gram_partial      : split-K partial Gram + norms  (2048 blocks, det.)
//   2b) gram_reduce_sm   : reduce 16 partials + softmax  (128 blocks)
//   3) attn_v            : out2 = attn @ v        (overwrites q region of ws)
//   4) dwconv3x3         : depthwise 3x3          (overwrites k region of ws)
//   5) gemm1x1_wmma      : out  = w_proj @ dw     (M=192, K=192, N=16384 / b)
//
// Workspace: qkv 8*576*16384 f32 (302MB) + attn 18432 f32 + gpart 344064 f32.
// ---------------------------------------------------------------------------

typedef float v2f __attribute__((ext_vector_type(2)));
typedef float v8f __attribute__((ext_vector_type(8)));

#define HW    16384
#define KDIM  192
#define LDA   196   // 16xK A-panel LDS stride: 196 % 64 == 4 -> conflict-free
#define LDQ   260   // gram tile row stride:    260 % 64 == 4 -> conflict-free
#define NCHUNK 16   // split-K factor for the Gram stage (1024 n per block)
#define GPART  168  // 144 Gram + 12 qnorm2 + 12 knorm2 partials per block

// Y[b][m][n] = sum_c W[m][c] * X[b][c][n], K=192, N=16384.
// Wave tile 16(M) x 64(N); block = 8 waves sharing one 16x192 A panel in LDS.
__global__ __launch_bounds__(256) void gemm1x1_wmma(
    const float* __restrict__ W,          // [M][192]
    const float* __restrict__ X,          // + b*xBatchStride -> [192][16384]
    float*       __restrict__ Y,          // + b*yBatchStride -> [M][16384]
    int M, long xBatchStride, long yBatchStride)
{
    __shared__ float lA[16 * LDA];

    const int tid   = threadIdx.x;
    const int wave  = tid >> 5;
    const int lane  = tid & 31;
    const int mBase = blockIdx.y * 16;
    const long b    = blockIdx.z;

    const float* __restrict__ Xb = X + b * xBatchStride;
    float*       __restrict__ Yb = Y + b * yBatchStride;

    // Stage A panel (16 x 192) into LDS, shared by all 8 waves.
    for (int i = tid; i < 16 * KDIM; i += 256) {
        const int r = i / KDIM, c = i % KDIM;
        lA[r * LDA + c] = W[(long)(mBase + r) * KDIM + c];
    }
    __syncthreads();

    const int n0   = blockIdx.x * 512 + wave * 64;  // wave's N origin
    const int nl   = lane & 15;
    const int koff = (lane >> 4) << 1;              // K sub-split: 0 or 2
    const int rowA = nl;                            // A row for this lane

    v8f acc0 = {}, acc1 = {}, acc2 = {}, acc3 = {};

    for (int kk = 0; kk < KDIM; kk += 4) {
        // A operand: VGPR0 = A[row][kk+koff], VGPR1 = A[row][kk+koff+1]
        const float2 af = *(const float2*)(&lA[rowA * LDA + kk + koff]);
        v2f a; a.x = af.x; a.y = af.y;

        const float* xr0 = Xb + (long)(kk + koff) * HW + n0 + nl;
        const float* xr1 = xr0 + HW;
        if (kk + 8 < KDIM) __builtin_prefetch(xr0 + 8L * HW, 0, 1);

        v2f b0; b0.x = xr0[0];  b0.y = xr1[0];
        acc0 = __builtin_amdgcn_wmma_f32_16x16x4_f32(false, a, false, b0,
                                                     (short)0, acc0, false, false);
        v2f b1; b1.x = xr0[16]; b1.y = xr1[16];
        acc1 = __builtin_amdgcn_wmma_f32_16x16x4_f32(false, a, false, b1,
                                                     (short)0, acc1, false, false);
        v2f b2; b2.x = xr0[32]; b2.y = xr1[32];
        acc2 = __builtin_amdgcn_wmma_f32_16x16x4_f32(false, a, false, b2,
                                                     (short)0, acc2, false, false);
        v2f b3; b3.x = xr0[48]; b3.y = xr1[48];
        acc3 = __builtin_amdgcn_wmma_f32_16x16x4_f32(false, a, false, b3,
                                                     (short)0, acc3, false, false);
    }

    // D layout: VGPR i, lanes 0-15 -> row i, lanes 16-31 -> row i+8, col=lane&15
    const int rowOff = (lane >> 4) * 8;
    for (int i = 0; i < 8; ++i) {
        float* yr = Yb + (long)(mBase + i + rowOff) * HW + n0 + nl;
        yr[0]  = acc0[i];
        yr[16] = acc1[i];
        yr[32] = acc2[i];
        yr[48] = acc3[i];
    }
}

// Pass 1 of the Gram stage: block (chunk, h, b) reduces 1024 spatial positions
// into 144 Gram partials + 12 q-norm^2 + 12 k-norm^2. Deterministic split-K
// (no float atomics): partials land in gpart[(b*16+h)][chunk][168].
__global__ __launch_bounds__(256) void gram_partial(
    const float* __restrict__ qkv,        // [b][576][16384]
    float*       __restrict__ gpart)      // [128][NCHUNK][GPART]
{
    const int chunk = blockIdx.x;
    const int h     = blockIdx.y;
    const int b     = blockIdx.z;
    const float* __restrict__ q = qkv + ((long)b * 576 + h * 12) * HW;
    const float* __restrict__ k = q + 192L * HW;

    __shared__ float lq[12 * LDQ];
    __shared__ float lk[12 * LDQ];

    const int tid = threadIdx.x;
    const int c = tid / 12, d = tid % 12;   // valid for tid < 144
    float acc = 0.f;

    const int nEnd = (chunk + 1) * (HW / NCHUNK);
    for (int nBase = chunk * (HW / NCHUNK); nBase < nEnd; nBase += 256) {
        __syncthreads();
        #pragma unroll
        for (int r = 0; r < 12; ++r) {
            lq[r * LDQ + tid] = q[(long)r * HW + nBase + tid];
            lk[r * LDQ + tid] = k[(long)r * HW + nBase + tid];
        }
        __syncthreads();
        if (tid < 144) {
            for (int j = 0; j < 256; ++j) acc += lq[c * LDQ + j] * lk[d * LDQ + j];
        } else if (tid < 156) {
            const int cc = tid - 144;
            for (int j = 0; j < 256; ++j) { float v = lq[cc * LDQ + j]; acc += v * v; }
        } else if (tid < 168) {
            const int cc = tid - 156;
            for (int j = 0; j < 256; ++j) { float v = lk[cc * LDQ + j]; acc += v * v; }
        }
    }
    if (tid < GPART)
        gpart[(((long)b * 16 + h) * NCHUNK + chunk) * GPART + tid] = acc;
}

// Pass 2: sum the NCHUNK partials (fixed order -> deterministic), apply the
// 1/(|q||k|) normalization (eps clamp) and temperature, row-softmax over 12.
__global__ __launch_bounds__(256) void gram_reduce_softmax(
    const float* __restrict__ gpart,
    const float* __restrict__ temperature,// [16]
    float*       __restrict__ attn)       // [b][16][12][12]
{
    const int b = blockIdx.x >> 4;
    const int h = blockIdx.x & 15;
    __shared__ float sG[144];
    __shared__ float sqn[12], skn[12];

    const int tid = threadIdx.x;
    if (tid < GPART) {
        const float* p = gpart + ((long)b * 16 + h) * NCHUNK * GPART + tid;
        float s = 0.f;
        #pragma unroll
        for (int c = 0; c < NCHUNK; ++c) s += p[c * GPART];
        if (tid < 144)      sG[tid]        = s;
        else if (tid < 156) sqn[tid - 144] = sqrtf(s);
        else                skn[tid - 156] = sqrtf(s);
    }
    __syncthreads();

    if (tid < 12) {
        const float t  = temperature[h];
        const float qn = fmaxf(sqn[tid], 1e-12f);
        float row[12];
        float mx = -1e30f;
        #pragma unroll
        for (int j = 0; j < 12; ++j) {
            const float kn = fmaxf(skn[j], 1e-12f);
            row[j] = sG[tid * 12 + j] / (qn * kn) * t;
            mx = fmaxf(mx, row[j]);
        }
        float s = 0.f;
        #pragma unroll
        for (int j = 0; j < 12; ++j) { row[j] = expf(row[j] - mx); s += row[j]; }
        const float inv = 1.f / s;
        float* arow = attn + (((long)b * 16 + h) * 12 + tid) * 12;
        #pragma unroll
        for (int j = 0; j < 12; ++j) arow[j] = row[j] * inv;
    }
}

// out2[b][h*12+c][n] = sum_d attn[b][h][c][d] * v[b][h*12+d][n]
// Writes over the (already consumed) q region of the qkv buffer.
__global__ __launch_bounds__(256) void attn_v(
    float* __restrict__ qkv, const float* __restrict__ attn)
{
    const int b = blockIdx.z, h = blockIdx.y;
    const int n = blockIdx.x * 256 + threadIdx.x;

    __shared__ float sA[144];
    if (threadIdx.x < 144)
        sA[threadIdx.x] = attn[((long)b * 16 + h) * 144 + threadIdx.x];
    __syncthreads();

    const float* __restrict__ v = qkv + ((long)b * 576 + 384 + h * 12) * HW + n;
    float*       __restrict__ o = qkv + ((long)b * 576 +       h * 12) * HW + n;

    float vv[12];
    #pragma unroll
    for (int d = 0; d < 12; ++d) vv[d] = v[(long)d * HW];
    #pragma unroll
    for (int c = 0; c < 12; ++c) {
        float s = 0.f;
        #pragma unroll
        for (int d = 0; d < 12; ++d) s += sA[c * 12 + d] * vv[d];
        o[(long)c * HW] = s;
    }
}

// Depthwise 3x3, pad 1 (cross-correlation, as XLA conv). Reads q region,
// writes k region of the qkv buffer.
__global__ __launch_bounds__(256) void dwconv3x3(
    float* __restrict__ qkv, const float* __restrict__ wdw)
{
    const int b = blockIdx.z, c = blockIdx.y;
    const int n = blockIdx.x * 256 + threadIdx.x;
    const int y = n >> 7, x = n & 127;

    const float* __restrict__ src = qkv + ((long)b * 576 + c) * HW;
    float wv[9];
    #pragma unroll
    for (int i = 0; i < 9; ++i) wv[i] = wdw[c * 9 + i];

    float s = 0.f;
    #pragma unroll
    for (int dy = -1; dy <= 1; ++dy) {
        const int yy = y + dy;
        if (yy < 0 || yy > 127) continue;
        #pragma unroll
        for (int dx = -1; dx <= 1; ++dx) {
            const int xx = x + dx;
            if (xx < 0 || xx > 127) continue;
            s += src[yy * 128 + xx] * wv[(dy + 1) * 3 + (dx + 1)];
        }
    }
    qkv[((long)b * 576 + 192 + c) * HW + n] = s;
}

extern "C" void kernel_launch(void* const* d_in, const int* in_sizes, int n_in,
                              void* d_out, int out_size, void* d_ws, size_t ws_size,
                              hipStream_t stream) {
    (void)in_sizes; (void)n_in; (void)out_size; (void)ws_size;
    const float* x      = (const float*)d_in[0];   // [8][192][128][128]
    const float* w_qkv  = (const float*)d_in[1];   // [576][192]
    const float* w_dw   = (const float*)d_in[2];   // [192][1][3][3]
    const float* w_proj = (const float*)d_in[3];   // [192][192]
    const float* temp   = (const float*)d_in[4];   // [16]
    float* out = (float*)d_out;                    // [8][192][128][128]

    float* qkv   = (float*)d_ws;                         // 8*576*16384 f32
    float* attn  = qkv + (size_t)8 * 576 * HW;           // 8*16*144 f32
    float* gpart = attn + (size_t)8 * 16 * 144;          // 128*16*168 f32

    // 1) qkv projection (WMMA GEMM)
    gemm1x1_wmma<<<dim3(32, 36, 8), 256, 0, stream>>>(
        w_qkv, x, qkv, 576, 192L * HW, 576L * HW);
    // 2) split-K Gram partials (deterministic, 2048 blocks)
    gram_partial<<<dim3(NCHUNK, 16, 8), 256, 0, stream>>>(qkv, gpart);
    // 2b) reduce + normalize + softmax
    gram_reduce_softmax<<<128, 256, 0, stream>>>(gpart, temp, attn);
    // 3) attn @ v  (q region reused as output)
    attn_v<<<dim3(64, 16, 8), 256, 0, stream>>>(qkv, attn);
    // 4) depthwise 3x3 (k region reused as output)
    dwconv3x3<<<dim3(64, 192, 8), 256, 0, stream>>>(qkv, w_dw);
    // 5) output projection (WMMA GEMM)
    gemm1x1_wmma<<<dim3(32, 12, 8), 256, 0, stream>>>(
        w_proj, qkv + 192L * HW, out, 192, 576L * HW, 192L * HW);
}